// CapsuleTransformConv_24369644438072
// MI455X (gfx1250) — compile-verified
//
#include <hip/hip_runtime.h>

// CapsuleTransformConv on gfx1250 (MI455X).
// 288 independent GEMMs: M=784 (b,oh,ow), N=512 (filter*atom), K=16 (in_atom), fp32.
// Memory-bound by the 462 MB fp32 output -> fp32 V_WMMA_F32_16X16X4_F32 + NT stores.
// Column loop fully unrolled (4 chunks/wave) so there is no divergence-masked
// loop: EXEC is structurally all-1s at every WMMA and the scheduler can hoist
// the next chunk's B-loads under the current chunk's WMMAs.

typedef __attribute__((ext_vector_type(2))) float v2f;
typedef __attribute__((ext_vector_type(8))) float v8f;

#define CIN      32
#define IN_ATOM  16
#define NCAP     288            // K*K*CIN
#define NCOLS    512            // FILTER*ATOM
#define OHW      14

__global__ __launch_bounds__(256)
void capsule_votes_wmma(const float* __restrict__ x,
                        const float* __restrict__ mat,
                        float* __restrict__ out) {
    const int n     = blockIdx.y;          // capsule patch index 0..287
    const int mtile = blockIdx.x;          // 0..48, 16 output positions each
    const int lane  = threadIdx.x & 31;
    const int wave  = threadIdx.x >> 5;    // 0..7
    const int half  = lane >> 4;           // lane half selects K sub-pairs
    const int l15   = lane & 15;

    // n -> (ki, kj, c)
    const int c  = n & 31;
    const int kj = (n >> 5) % 3;
    const int ki = (n >> 5) / 3;

    // This lane's A-matrix row: output position m = mtile*16 + l15  (always < 784)
    const int m  = mtile * 16 + l15;
    const int b  = m / 196;
    const int r0 = m - b * 196;
    const int oh = r0 / OHW;
    const int ow = r0 - oh * OHW;
    const int h  = oh + ki;
    const int w  = ow + kj;

    // x layout: [B,16,16,32,16] fp32
    const float* xrow = x + ((((b * 16 + h) * 16 + w) * CIN + c) << 4);

    // A fragment (16x4 fp32 per k-step): lanes 0-15 hold K={4k,4k+1},
    // lanes 16-31 hold K={4k+2,4k+3}.  Four k-steps cover K=16.
    v2f afrag[4];
#pragma unroll
    for (int kk = 0; kk < 4; ++kk) {
        const int a0 = kk * 4 + half * 2;
        afrag[kk] = *(const v2f*)(xrow + a0);     // contiguous float2, 8B aligned
    }

    // mat layout: [288, 16, 512] fp32
    const float* bbase = mat + (size_t)n * IN_ATOM * NCOLS;
    const size_t orow_stride = (size_t)NCAP * NCOLS;     // 147456

    // 32 sixteen-column chunks / 8 waves = exactly 4 chunks per wave.
#pragma unroll
    for (int i = 0; i < 4; ++i) {
        const int chunk = wave + i * 8;           // wave-uniform, compile-time trip count
        const int col   = chunk * 16 + l15;
        const float* bcol = bbase + col;

        v8f acc = {0.f, 0.f, 0.f, 0.f, 0.f, 0.f, 0.f, 0.f};
#pragma unroll
        for (int kk = 0; kk < 4; ++kk) {
            const int a0 = kk * 4 + half * 2;
            v2f bfrag;                            // B 4x16: rows striped across lanes
            bfrag.x = bcol[(size_t)a0 * NCOLS];
            bfrag.y = bcol[(size_t)(a0 + 1) * NCOLS];
            acc = __builtin_amdgcn_wmma_f32_16x16x4_f32(
                /*neg_a=*/false, afrag[kk],
                /*neg_b=*/false, bfrag,
                /*c_mod=*/(short)0, acc,
                /*reuse_a=*/false, /*reuse_b=*/false);
        }

        // D 16x16 fp32: VGPR r -> row (r + 8*half), col = l15 within chunk.
        // out layout: [784, 288, 512]; streaming store, keep L2 for the matrix.
        float* obase = out + ((size_t)(mtile * 16 + half * 8) * NCAP + n) * NCOLS + col;
#pragma unroll
        for (int r = 0; r < 8; ++r) {
            __builtin_nontemporal_store(acc[r], obase + (size_t)r * orow_stride);
        }
    }
}

extern "C" void kernel_launch(void* const* d_in, const int* in_sizes, int n_in,
                              void* d_out, int out_size, void* d_ws, size_t ws_size,
                              hipStream_t stream) {
    const float* x   = (const float*)d_in[0];   // [4,16,16,32,16] fp32
    const float* mat = (const float*)d_in[1];   // [288,16,512] fp32
    float*       out = (float*)d_out;           // [4,14,14,288,32,16] fp32

    dim3 grid(49, NCAP);   // 49 m-tiles x 288 capsule patches
    capsule_votes_wmma<<<grid, 256, 0, stream>>>(x, mat, out);
}